// SymplecticGNN_51049981280667
// MI455X (gfx1250) — compile-verified
//
#include <hip/hip_runtime.h>
#include <hip/hip_bf16.h>

typedef __attribute__((ext_vector_type(2))) float v2f;
typedef __attribute__((ext_vector_type(8))) float v8f;

#define HID 128

__device__ __forceinline__ void atomAddF(float* p, float v) {
    unsafeAtomicAdd(p, v);   // -> global_atomic_add_f32 (no CAS loop)
}

// lds byte offset of a generic pointer into LDS (addrspacecast -> ptrtoint)
__device__ __forceinline__ unsigned ldsOffset(const void* p) {
    return (unsigned)(unsigned long long)(__attribute__((address_space(3))) const char*)p;
}

// ---------------- init: deg=1 (self loop), pooled=0, cnt=0 ----------------
__global__ __launch_bounds__(256) void initK(float* deg, float* pooled, float* cnt,
                                             int N, int G) {
    int id = blockIdx.x * 256 + threadIdx.x;
    if (id < N) deg[id] = 1.0f;
    if (id < G * HID) pooled[id] = 0.0f;
    if (id < G) cnt[id] = 0.0f;
}

// ---------------- degree accumulation on dst ----------------
__global__ __launch_bounds__(256) void degK(const int* __restrict__ dst,
                                            float* __restrict__ deg, int E) {
    int e = blockIdx.x * 256 + threadIdx.x;
    if (e < E) atomAddF(&deg[dst[e]], 1.0f);
}

// ---------------- dinv = rsqrt(deg), self_norm = 1/deg ----------------
__global__ __launch_bounds__(256) void dinvK(const float* __restrict__ deg,
                                             float* __restrict__ dinv,
                                             float* __restrict__ selfn, int N) {
    int n = blockIdx.x * 256 + threadIdx.x;
    if (n < N) {
        float r = rsqrtf(deg[n]);
        dinv[n] = r;
        selfn[n] = r * r;
    }
}

// ---------------- edge norm = dinv[src]*dinv[dst] ----------------
__global__ __launch_bounds__(256) void normK(const int* __restrict__ src,
                                             const int* __restrict__ dst,
                                             const float* __restrict__ dinv,
                                             float* __restrict__ nrm, int E) {
    int e = blockIdx.x * 256 + threadIdx.x;
    if (e < E) nrm[e] = dinv[src[e]] * dinv[dst[e]];
}

// ---------------- encoder: h = relu(x @ encW + encB), NF=4 ----------------
__global__ __launch_bounds__(256) void encK(const float* __restrict__ x,
                                            const float* __restrict__ W,
                                            const float* __restrict__ B,
                                            float* __restrict__ h, int N) {
    int id = blockIdx.x * 256 + threadIdx.x;
    if (id >= N * HID) return;
    int n = id >> 7, c = id & 127;
    const float* xr = x + (size_t)n * 4;
    float s = B[c];
#pragma unroll
    for (int f = 0; f < 4; ++f) s += xr[f] * W[f * HID + c];
    h[id] = fmaxf(s, 0.0f);
}

// ---------------- fp32 WMMA GEMM: [N,128] @ [128,128] ----------------
// MODE 0: out = A@W (hw);  agg = out*selfn + bias   (fused conv epilogue)
// MODE 1: out += relu(A)@W (relu applied on A-fragment read)
#define LDSP 132   // row stride: 16B-aligned rows, conflict-free banks (4r mod 64)
template <int MODE>
__global__ __launch_bounds__(256) void gemmK(const float* __restrict__ A,
                                             const float* __restrict__ W,
                                             float* __restrict__ out,
                                             float* __restrict__ agg,
                                             const float* __restrict__ selfn,
                                             const float* __restrict__ bias, int N) {
    __shared__ float As[64][LDSP];
    const int t = threadIdx.x;
    const int rowBase = blockIdx.x * 64;
    const bool full = (rowBase + 64 <= N);

    if (full) {
        // Async global->LDS staging: 16B per lane, no VGPR round-trip (ASYNCcnt)
#pragma unroll
        for (int i = 0; i < 8; ++i) {
            int row = (t >> 5) + i * 8;
            int col = (t & 31) * 4;
            const float* ga = A + (size_t)(rowBase + row) * HID + col;
            unsigned lo = ldsOffset(&As[row][col]);
            asm volatile("global_load_async_to_lds_b128 %0, %1, off"
                         :: "v"(lo), "v"(ga) : "memory");
        }
        asm volatile("s_wait_asynccnt 0x0" ::: "memory");
    } else {
        // Tail block (1 of ~1563): manual staging with zero fill
#pragma unroll
        for (int i = 0; i < 8; ++i) {
            int row = (t >> 5) + i * 8;
            int col = (t & 31) * 4;
            int g = rowBase + row;
            float4 v = make_float4(0.f, 0.f, 0.f, 0.f);
            if (g < N) v = *(const float4*)(A + (size_t)g * HID + col);
            *(float4*)(&As[row][col]) = v;
        }
    }
    __syncthreads();

    const int wave = t >> 5, lane = t & 31;
    const int rt = wave >> 1;          // row tile 0..3 (16 rows each)
    const int ch = wave & 1;           // column half 0..1 (64 cols each)
    const int lrow = rt * 16 + (lane & 15);
    const int khalf = (lane >> 4) * 2; // lanes 0-15 -> K{0,1}, 16-31 -> K{2,3}
    const int cb = ch * 64 + (lane & 15);

    v8f acc[4] = {};
    for (int kk = 0; kk < 32; ++kk) {
        const int k = kk * 4 + khalf;
        v2f a = *(const v2f*)(&As[lrow][k]);  // ds_load_b64, conflict-free
        if (MODE == 1) {                       // relu(A) fused into fragment read
            a.x = fmaxf(a.x, 0.f);
            a.y = fmaxf(a.y, 0.f);
        }
#pragma unroll
        for (int ct = 0; ct < 4; ++ct) {
            const int col = cb + ct * 16;
            v2f b;
            b.x = W[(size_t)k * HID + col];
            b.y = W[(size_t)(k + 1) * HID + col];
            acc[ct] = __builtin_amdgcn_wmma_f32_16x16x4_f32(
                false, a, false, b, (short)0, acc[ct], false, false);
        }
    }

    // C layout: lane 0-15 -> N=lane, VGPR r -> M=r; lanes 16-31 -> M=r+8
    const int rWr = rowBase + rt * 16 + 8 * (lane >> 4);
    const int cWr = ch * 64 + (lane & 15);

    if (full) {                       // uniform fast path: no exec-mask churn
        if (MODE == 0) {
            float bv[4], sn[8];
#pragma unroll
            for (int ct = 0; ct < 4; ++ct) bv[ct] = bias[cWr + ct * 16];
#pragma unroll
            for (int r = 0; r < 8; ++r) sn[r] = selfn[rWr + r];
#pragma unroll
            for (int ct = 0; ct < 4; ++ct)
#pragma unroll
                for (int r = 0; r < 8; ++r) {
                    size_t idx = (size_t)(rWr + r) * HID + cWr + ct * 16;
                    float v = acc[ct][r];
                    out[idx] = v;
                    agg[idx] = v * sn[r] + bv[ct];
                }
        } else {
#pragma unroll
            for (int ct = 0; ct < 4; ++ct)
#pragma unroll
                for (int r = 0; r < 8; ++r) {
                    size_t idx = (size_t)(rWr + r) * HID + cWr + ct * 16;
                    out[idx] += acc[ct][r];
                }
        }
    } else {                          // tail block only
#pragma unroll
        for (int ct = 0; ct < 4; ++ct)
#pragma unroll
            for (int r = 0; r < 8; ++r) {
                int row = rWr + r;
                if (row < N) {
                    int col = cWr + ct * 16;
                    float v = acc[ct][r];
                    size_t idx = (size_t)row * HID + col;
                    if (MODE == 0) {
                        out[idx] = v;
                        agg[idx] = v * selfn[row] + bias[col];
                    } else {
                        out[idx] += v;
                    }
                }
            }
    }
}

// ---------------- edge messages: agg[dst] += norm * hw[src], wave-per-edge ----
__global__ __launch_bounds__(256) void edgeK(const int* __restrict__ src,
                                             const int* __restrict__ dst,
                                             const float* __restrict__ nrm,
                                             const float* __restrict__ hw,
                                             float* __restrict__ agg, int E) {
    int e = blockIdx.x * 8 + (threadIdx.x >> 5);
    if (e >= E) return;
    int lane = threadIdx.x & 31;
    int s = src[e], d = dst[e];
    float nv = nrm[e];
    const float4 m = *(const float4*)(hw + (size_t)s * HID + lane * 4);
    float* ap = agg + (size_t)d * HID + lane * 4;
    atomAddF(ap + 0, m.x * nv);
    atomAddF(ap + 1, m.y * nv);
    atomAddF(ap + 2, m.z * nv);
    atomAddF(ap + 3, m.w * nv);
}

// ---------------- global mean pool (scatter phase), wave-per-node ------------
__global__ __launch_bounds__(256) void poolK(const float* __restrict__ h,
                                             const int* __restrict__ batch,
                                             float* __restrict__ pooled,
                                             float* __restrict__ cnt, int N) {
    int n = blockIdx.x * 8 + (threadIdx.x >> 5);
    if (n >= N) return;
    int lane = threadIdx.x & 31;
    int b = batch[n];
    const float4 v = *(const float4*)(h + (size_t)n * HID + lane * 4);
    float* pp = pooled + (size_t)b * HID + lane * 4;
    atomAddF(pp + 0, v.x);
    atomAddF(pp + 1, v.y);
    atomAddF(pp + 2, v.z);
    atomAddF(pp + 3, v.w);
    if (lane == 0) atomAddF(&cnt[b], 1.0f);
}

// ---------------- pooled -> mean (in place) ----------------
__global__ __launch_bounds__(256) void meanK(float* __restrict__ pooled,
                                             const float* __restrict__ cnt, int G) {
    int id = blockIdx.x * 256 + threadIdx.x;
    if (id < G * HID) pooled[id] /= fmaxf(cnt[id >> 7], 1.0f);
}

// ---------------- decoder: relu(mean@W1+B1)@W2+B2, one block per graph -------
__global__ __launch_bounds__(64) void decK(const float* __restrict__ mean,
                                           const float* __restrict__ W1,
                                           const float* __restrict__ B1,
                                           const float* __restrict__ W2,
                                           const float* __restrict__ B2,
                                           float* __restrict__ out) {
    __shared__ float m[HID];
    __shared__ float t1[64];
    int g = blockIdx.x, t = threadIdx.x;
    m[t] = mean[(size_t)g * HID + t];
    m[t + 64] = mean[(size_t)g * HID + t + 64];
    __syncthreads();
    float s = B1[t];
#pragma unroll 8
    for (int k = 0; k < HID; ++k) s += m[k] * W1[(size_t)k * 64 + t];
    t1[t] = fmaxf(s, 0.0f);
    __syncthreads();
    if (t < 4) {
        float o = B2[t];
#pragma unroll 8
        for (int j = 0; j < 64; ++j) o += t1[j] * W2[j * 4 + t];
        out[g * 4 + t] = o;
    }
}

extern "C" void kernel_launch(void* const* d_in, const int* in_sizes, int n_in,
                              void* d_out, int out_size, void* d_ws, size_t ws_size,
                              hipStream_t stream) {
    const float* x     = (const float*)d_in[0];
    const int*   eidx  = (const int*)d_in[1];
    const int*   batch = (const int*)d_in[2];
    const float* encW  = (const float*)d_in[3];
    const float* encB  = (const float*)d_in[4];
    const float* convW = (const float*)d_in[5];
    const float* convB = (const float*)d_in[6];
    const float* sympW = (const float*)d_in[7];
    const float* decW1 = (const float*)d_in[8];
    const float* decB1 = (const float*)d_in[9];
    const float* decW2 = (const float*)d_in[10];
    const float* decB2 = (const float*)d_in[11];

    const int N  = in_sizes[0] / 4;       // NF = 4
    const int E  = in_sizes[1] / 2;
    const int G  = out_size / 4;
    const int NL = in_sizes[6] / HID;     // convB is (NL, HID)

    const int* src = eidx;
    const int* dst = eidx + E;

    float* ws     = (float*)d_ws;
    float* h      = ws;
    float* hw     = h      + (size_t)N * HID;
    float* agg    = hw     + (size_t)N * HID;
    float* deg    = agg    + (size_t)N * HID;
    float* dinv   = deg    + N;
    float* selfn  = dinv   + N;
    float* nrm    = selfn  + N;
    float* pooled = nrm    + E;
    float* cnt    = pooled + (size_t)G * HID;

    dim3 b256(256);
    initK<<<(N + 255) / 256, b256, 0, stream>>>(deg, pooled, cnt, N, G);
    degK<<<(E + 255) / 256, b256, 0, stream>>>(dst, deg, E);
    dinvK<<<(N + 255) / 256, b256, 0, stream>>>(deg, dinv, selfn, N);
    normK<<<(E + 255) / 256, b256, 0, stream>>>(src, dst, dinv, nrm, E);
    encK<<<(N * HID + 255) / 256, b256, 0, stream>>>(x, encW, encB, h, N);

    const int gblocks = (N + 63) / 64;
    for (int i = 0; i < NL; ++i) {
        gemmK<0><<<gblocks, b256, 0, stream>>>(
            h, convW + (size_t)i * HID * HID, hw, agg, selfn, convB + (size_t)i * HID, N);
        edgeK<<<(E + 7) / 8, b256, 0, stream>>>(src, dst, nrm, hw, agg, E);
        gemmK<1><<<gblocks, b256, 0, stream>>>(
            agg, sympW + (size_t)i * HID * HID, h, nullptr, nullptr, nullptr, N);
    }

    poolK<<<(N + 7) / 8, b256, 0, stream>>>(h, batch, pooled, cnt, N);
    meanK<<<(G * HID + 255) / 256, b256, 0, stream>>>(pooled, cnt, G);
    decK<<<G, 64, 0, stream>>>(pooled, decW1, decB1, decW2, decB2, (float*)d_out);
}